// Decoder_78091095376336
// MI455X (gfx1250) — compile-verified
//
#include <hip/hip_runtime.h>
#include <math.h>

typedef __attribute__((ext_vector_type(2))) float v2f;
typedef __attribute__((ext_vector_type(8))) float v8f;
typedef __attribute__((ext_vector_type(4))) unsigned int u32x4;
typedef __attribute__((ext_vector_type(8))) int i32x8;
typedef __attribute__((ext_vector_type(4))) int i32x4;

#define BATCH 512
#define EDIM  1024
#define MLEN  50
#define VOC   50000

#if defined(__has_builtin)
#  if __has_builtin(__builtin_amdgcn_tensor_load_to_lds) && \
      __has_builtin(__builtin_amdgcn_s_wait_tensorcnt)
#    define USE_TDM 1
#  endif
#endif
#ifndef USE_TDM
#  define USE_TDM 0
#endif

// Projection GEMM tiling: K-chunk 64, pad codes: interval 5 (=64 DW), amount 3 (=4 DW)
#define BKP   64
#define LDSP  (BKP + 4)
// Gate GEMM tiling: K-chunk 32, pad codes: interval 4 (=32 DW), amount 3 (=4 DW)
#define BKG   32
#define LDSG  (BKG + 4)

static __device__ __forceinline__ float sigmoidf_(float x) {
    return 1.0f / (1.0f + expf(-x));
}

#if USE_TDM
// Generic 2D/3D fp32 TDM tile load into LDS with per-row DWORD padding.
//   bk   : contiguous elements per row (dim0), rows: dim1, nz: dim2 (0 => 2D)
//   row_stride / z_stride in elements. LDS rows land padded (pad_amount DWORDs).
static __device__ __forceinline__ void tdm_load_tile(const float* gptr,
                                                     unsigned lds_byte,
                                                     int bk, int rows, int nz,
                                                     long long row_stride,
                                                     long long z_stride,
                                                     int pad_interval_code,
                                                     int pad_amount_code) {
    unsigned long long ga = (unsigned long long)(uintptr_t)gptr;
    // D# group 0: count=1 | lds_addr | global_addr[56:0] | type=2
    u32x4 g0 = { 1u,
                 lds_byte,
                 (unsigned)ga,
                 (unsigned)((ga >> 32) & 0x01FFFFFFu) | (2u << 30) };
    // D# group 1 (bit-packed per ISA §8.4)
    i32x8 g1 = {
        (int)((2u << 16) | (1u << 20) |
              ((unsigned)pad_interval_code << 22) |
              ((unsigned)pad_amount_code  << 25)),                 // data_size=4B, pad
        (int)((unsigned)(bk & 0xFFFF) << 16),                      // tensor_dim0[15:0]
        (int)(((unsigned)rows & 0xFFFF) << 16),                    // tensor_dim1[15:0]
        (int)((unsigned)(bk & 0xFFFF) << 16),                      // tile_dim0
        (int)(((unsigned)rows & 0xFFFF) |
              (((unsigned)nz & 0xFFFF) << 16)),                    // tile_dim1 | tile_dim2
        (int)(row_stride & 0xFFFFFFFFll),                          // dim0_stride[31:0]
        (int)(((row_stride >> 32) & 0xFFFFll) |
              ((z_stride & 0xFFFFll) << 16)),                      // dim0_str[47:32]|dim1_str[15:0]
        (int)((z_stride >> 16) & 0xFFFFFFFFll) };                  // dim1_stride[47:16]
    // D# group 2: tensor_dim2 (z length); rest zero (no iterate, no dim3)
    i32x4 g2 = { nz, 0, 0, 0 };
    i32x4 g3 = { 0, 0, 0, 0 };
    i32x8 z8 = { 0, 0, 0, 0, 0, 0, 0, 0 };
    __builtin_amdgcn_tensor_load_to_lds(g0, g1, g2, g3, z8, 0);
}

static __device__ __forceinline__ unsigned lds_off(const void* p) {
    return (unsigned)(uintptr_t)p;   // LDS aperture: low 32 bits == byte offset
}
#endif

// ---------------------------------------------------------------------------
// Kernel A: attention energies -> masked softmax -> context; embedding gather.
// ---------------------------------------------------------------------------
__global__ void __launch_bounds__(256)
attn_ctx_kernel(const int* __restrict__ tgt,
                const float* __restrict__ enc,
                const float* __restrict__ h0,
                const unsigned char* __restrict__ pad,
                const float* __restrict__ emb,
                float* __restrict__ words,
                float* __restrict__ ctx)
{
    const int b    = blockIdx.x;
    const int tid  = threadIdx.x;
    const int lane = tid & 31;
    const int wave = tid >> 5;

    __shared__ float sh_e[MLEN];

    const float* hrow = h0 + (size_t)b * EDIM;
    const float* erow = enc + (size_t)b * MLEN * EDIM;

    for (int l = wave; l < MLEN; l += 8) {
        const float* el = erow + (size_t)l * EDIM;
        float p = 0.0f;
        for (int e = lane; e < EDIM; e += 32)
            p += hrow[e] * el[e];
        for (int off = 16; off > 0; off >>= 1)
            p += __shfl_down(p, off, 32);
        if (lane == 0) sh_e[l] = p;
    }
    __syncthreads();

    if (tid == 0) {
        float mx = sh_e[0];
        for (int l = 1; l < MLEN; ++l) mx = fmaxf(mx, sh_e[l]);
        float s = 0.0f;
        for (int l = 0; l < MLEN; ++l) { sh_e[l] = expf(sh_e[l] - mx); s += sh_e[l]; }
        float inv = 1.0f / s;
        float s2 = 0.0f;
        for (int l = 0; l < MLEN; ++l) {
            float v = pad[(size_t)b * MLEN + l] ? 0.0f : sh_e[l] * inv;
            sh_e[l] = v;
            s2 += v;
        }
        float inv2 = 1.0f / s2;
        for (int l = 0; l < MLEN; ++l) sh_e[l] *= inv2;
    }
    __syncthreads();

    const float* wrow = emb + (size_t)tgt[b] * EDIM;
    for (int e = tid; e < EDIM; e += 256) {
        float c = 0.0f;
        #pragma unroll 5
        for (int l = 0; l < MLEN; ++l)
            c += sh_e[l] * erow[(size_t)l * EDIM + e];
        ctx[(size_t)b * EDIM + e]   = c;
        words[(size_t)b * EDIM + e] = wrow[e];
    }
}

// ---------------------------------------------------------------------------
// Kernel B: gates = words@w_ih^T + ctx@w_hh^T + biases, fused LSTM cell.
// Block = 8 waves -> 128(M) x 16(h) x 4 gates. TDM-staged LDS tiles:
//   A_words/A_ctx as 2D tiles, w_ih/w_hh as 3D tiles (gate dim via dim1_stride).
// ---------------------------------------------------------------------------
__global__ void __launch_bounds__(256)
gates_lstm_kernel(const float* __restrict__ words,
                  const float* __restrict__ ctx,
                  const float* __restrict__ w_ih,
                  const float* __restrict__ w_hh,
                  const float* __restrict__ b_ih,
                  const float* __restrict__ b_hh,
                  float* __restrict__ h_out,
                  float* __restrict__ c_out)
{
    __shared__ float sAw[2][128][LDSG];
    __shared__ float sAc[2][128][LDSG];
    __shared__ float sBi[2][64][LDSG];    // 4 gates x 16 rows
    __shared__ float sBh[2][64][LDSG];

    const int tid  = threadIdx.x;
    const int lane = tid & 31;
    const int wave = tid >> 5;
    const int lr   = lane & 15;
    const int lb   = lane >> 4;
    const int kofs = 2 * lb;

    const int mblk  = blockIdx.x & 3;        // 4 m-blocks of 128
    const int hblk  = blockIdx.x >> 2;       // 64 hidden-tiles of 16
    const int mbase = mblk * 128;
    const int hc0   = hblk * 16;

    const float* Aw = words + (size_t)mbase * EDIM;
    const float* Ac = ctx   + (size_t)mbase * EDIM;
    const float* Bi = w_ih  + (size_t)hc0 * EDIM;   // 3D: +r*EDIM, +q*EDIM*EDIM
    const float* Bh = w_hh  + (size_t)hc0 * EDIM;

    v8f acc[4];
    #pragma unroll
    for (int q = 0; q < 4; ++q) acc[q] = (v8f)(0.0f);

    const long long ZSTR = (long long)EDIM * EDIM;

#if USE_TDM
    if      (wave == 0) tdm_load_tile(Aw, lds_off(&sAw[0][0][0]), BKG, 128, 0, EDIM, 0,    4, 3);
    else if (wave == 1) tdm_load_tile(Ac, lds_off(&sAc[0][0][0]), BKG, 128, 0, EDIM, 0,    4, 3);
    else if (wave == 2) tdm_load_tile(Bi, lds_off(&sBi[0][0][0]), BKG, 16,  4, EDIM, ZSTR, 4, 3);
    else if (wave == 3) tdm_load_tile(Bh, lds_off(&sBh[0][0][0]), BKG, 16,  4, EDIM, ZSTR, 4, 3);
    if (wave < 4) __builtin_amdgcn_s_wait_tensorcnt(0);
    __syncthreads();

    int buf = 0;
    for (int kc = 0; kc < EDIM; kc += BKG) {
        const int nxt = buf ^ 1;
        if ((kc + BKG) < EDIM) {
            const int kn = kc + BKG;
            if      (wave == 0) tdm_load_tile(Aw + kn, lds_off(&sAw[nxt][0][0]), BKG, 128, 0, EDIM, 0,    4, 3);
            else if (wave == 1) tdm_load_tile(Ac + kn, lds_off(&sAc[nxt][0][0]), BKG, 128, 0, EDIM, 0,    4, 3);
            else if (wave == 2) tdm_load_tile(Bi + kn, lds_off(&sBi[nxt][0][0]), BKG, 16,  4, EDIM, ZSTR, 4, 3);
            else if (wave == 3) tdm_load_tile(Bh + kn, lds_off(&sBh[nxt][0][0]), BKG, 16,  4, EDIM, ZSTR, 4, 3);
        }

        const float* aw = &sAw[buf][wave * 16 + lr][kofs];
        const float* ac = &sAc[buf][wave * 16 + lr][kofs];
        const float* bi = &sBi[buf][lr][kofs];
        const float* bh = &sBh[buf][lr][kofs];
        #pragma unroll
        for (int k = 0; k < BKG; k += 4) {
            const v2f a1 = *(const v2f*)(aw + k);
            const v2f a2 = *(const v2f*)(ac + k);
            #pragma unroll
            for (int q = 0; q < 4; ++q) {
                const v2f b1 = *(const v2f*)(bi + q * 16 * LDSG + k);
                acc[q] = __builtin_amdgcn_wmma_f32_16x16x4_f32(
                    false, a1, false, b1, (short)0, acc[q], false, false);
                const v2f b2 = *(const v2f*)(bh + q * 16 * LDSG + k);
                acc[q] = __builtin_amdgcn_wmma_f32_16x16x4_f32(
                    false, a2, false, b2, (short)0, acc[q], false, false);
            }
        }

        if (wave < 4) __builtin_amdgcn_s_wait_tensorcnt(0);
        __syncthreads();
        buf = nxt;
    }
#else
    for (int kc = 0; kc < EDIM; kc += BKG) {
        for (int idx = tid; idx < 128 * (BKG / 4); idx += 256) {
            const int row = idx / (BKG / 4);
            const int c4  = (idx % (BKG / 4)) * 4;
            *(float4*)&sAw[0][row][c4] = *(const float4*)(Aw + (size_t)row * EDIM + kc + c4);
            *(float4*)&sAc[0][row][c4] = *(const float4*)(Ac + (size_t)row * EDIM + kc + c4);
        }
        for (int idx = tid; idx < 64 * (BKG / 4); idx += 256) {
            const int row = idx / (BKG / 4);          // q*16 + r
            const int c4  = (idx % (BKG / 4)) * 4;
            const int q   = row >> 4;
            const int r   = row & 15;
            const size_t goff = ((size_t)q * EDIM + r) * EDIM + kc + c4;
            *(float4*)&sBi[0][row][c4] = *(const float4*)(Bi + goff);
            *(float4*)&sBh[0][row][c4] = *(const float4*)(Bh + goff);
        }
        __syncthreads();

        const float* aw = &sAw[0][wave * 16 + lr][kofs];
        const float* ac = &sAc[0][wave * 16 + lr][kofs];
        const float* bi = &sBi[0][lr][kofs];
        const float* bh = &sBh[0][lr][kofs];
        #pragma unroll
        for (int k = 0; k < BKG; k += 4) {
            const v2f a1 = *(const v2f*)(aw + k);
            const v2f a2 = *(const v2f*)(ac + k);
            #pragma unroll
            for (int q = 0; q < 4; ++q) {
                const v2f b1 = *(const v2f*)(bi + q * 16 * LDSG + k);
                acc[q] = __builtin_amdgcn_wmma_f32_16x16x4_f32(
                    false, a1, false, b1, (short)0, acc[q], false, false);
                const v2f b2 = *(const v2f*)(bh + q * 16 * LDSG + k);
                acc[q] = __builtin_amdgcn_wmma_f32_16x16x4_f32(
                    false, a2, false, b2, (short)0, acc[q], false, false);
            }
        }
        __syncthreads();
    }
#endif

    const int col = hc0 + lr;
    const float biasI = b_ih[0 * EDIM + col] + b_hh[0 * EDIM + col];
    const float biasF = b_ih[1 * EDIM + col] + b_hh[1 * EDIM + col];
    const float biasG = b_ih[2 * EDIM + col] + b_hh[2 * EDIM + col];
    const float biasO = b_ih[3 * EDIM + col] + b_hh[3 * EDIM + col];

    #pragma unroll
    for (int r = 0; r < 8; ++r) {
        const int m = mbase + wave * 16 + r + 8 * lb;
        const float cv = ctx[(size_t)m * EDIM + col];
        const float iv = sigmoidf_(acc[0][r] + biasI);
        const float fv = sigmoidf_(acc[1][r] + biasF);
        const float gv = tanhf    (acc[2][r] + biasG);
        const float ov = sigmoidf_(acc[3][r] + biasO);
        const float cn = fv * cv + iv * gv;
        const float hn = ov * tanhf(cn);
        h_out[(size_t)m * EDIM + col] = hn;
        c_out[(size_t)m * EDIM + col] = cn;
    }
}

// ---------------------------------------------------------------------------
// Kernel C: logits = h_new @ w_proj^T + b_proj.
// Block = 8 waves -> 128(M) x 16(N); K chunked by 64, TDM double-buffered LDS.
// ---------------------------------------------------------------------------
__global__ void __launch_bounds__(256)
proj_kernel(const float* __restrict__ hnew,
            const float* __restrict__ w_proj,
            const float* __restrict__ b_proj,
            float* __restrict__ logits)
{
    __shared__ float shA[2][128][LDSP];
    __shared__ float shB[2][16][LDSP];

    const int tid  = threadIdx.x;
    const int lane = tid & 31;
    const int wave = tid >> 5;
    const int lr   = lane & 15;
    const int lb   = lane >> 4;
    const int kofs = 2 * lb;

    const int mblk  = blockIdx.x & 3;     // m fastest -> w_proj rows reused in L2
    const int nblk  = blockIdx.x >> 2;
    const int mbase = mblk * 128;
    const int nbase = nblk * 16;

    const float* Ag = hnew   + (size_t)mbase * EDIM;
    const float* Bg = w_proj + (size_t)nbase * EDIM;

    v8f a0 = (v8f)(0.0f), a1 = (v8f)(0.0f), a2 = (v8f)(0.0f), a3 = (v8f)(0.0f);

#if USE_TDM
    if      (wave == 0) tdm_load_tile(Ag, lds_off(&shA[0][0][0]), BKP, 128, 0, EDIM, 0, 5, 3);
    else if (wave == 1) tdm_load_tile(Bg, lds_off(&shB[0][0][0]), BKP, 16,  0, EDIM, 0, 5, 3);
    if (wave < 2) __builtin_amdgcn_s_wait_tensorcnt(0);
    __syncthreads();

    int buf = 0;
    for (int kc = 0; kc < EDIM; kc += BKP) {
        const int nxt = buf ^ 1;
        if ((kc + BKP) < EDIM) {
            if      (wave == 0) tdm_load_tile(Ag + kc + BKP, lds_off(&shA[nxt][0][0]), BKP, 128, 0, EDIM, 0, 5, 3);
            else if (wave == 1) tdm_load_tile(Bg + kc + BKP, lds_off(&shB[nxt][0][0]), BKP, 16,  0, EDIM, 0, 5, 3);
        }

        const float* ar = &shA[buf][wave * 16 + lr][kofs];
        const float* br = &shB[buf][lr][kofs];
        #pragma unroll
        for (int k = 0; k < BKP; k += 16) {
            v2f av, bv;
            av = *(const v2f*)(ar + k + 0);
            bv = *(const v2f*)(br + k + 0);
            a0 = __builtin_amdgcn_wmma_f32_16x16x4_f32(false, av, false, bv, (short)0, a0, false, false);
            av = *(const v2f*)(ar + k + 4);
            bv = *(const v2f*)(br + k + 4);
            a1 = __builtin_amdgcn_wmma_f32_16x16x4_f32(false, av, false, bv, (short)0, a1, false, false);
            av = *(const v2f*)(ar + k + 8);
            bv = *(const v2f*)(br + k + 8);
            a2 = __builtin_amdgcn_wmma_f32_16x16x4_f32(false, av, false, bv, (short)0, a2, false, false);
            av = *(const v2f*)(ar + k + 12);
            bv = *(const v2f*)(br + k + 12);
            a3 = __builtin_amdgcn_wmma_f32_16x16x4_f32(false, av, false, bv, (short)0, a3, false, false);
        }

        if (wave < 2) __builtin_amdgcn_s_wait_tensorcnt(0);
        __syncthreads();
        buf = nxt;
    }
#else
    for (int kc = 0; kc < EDIM; kc += BKP) {
        for (int idx = tid; idx < 128 * (BKP / 4); idx += 256) {
            const int row = idx / (BKP / 4);
            const int c4  = (idx % (BKP / 4)) * 4;
            *(float4*)&shA[0][row][c4] =
                *(const float4*)(Ag + (size_t)row * EDIM + kc + c4);
        }
        for (int idx = tid; idx < 16 * (BKP / 4); idx += 256) {
            const int row = idx / (BKP / 4);
            const int c4  = (idx % (BKP / 4)) * 4;
            *(float4*)&shB[0][row][c4] =
                *(const float4*)(Bg + (size_t)row * EDIM + kc + c4);
        }
        __syncthreads();

        const float* ar = &shA[0][wave * 16 + lr][kofs];
        const float* br = &shB[0][lr][kofs];
        #pragma unroll
        for (int k = 0; k < BKP; k += 16) {
            v2f av, bv;
            av = *(const v2f*)(ar + k + 0);
            bv = *(const v2f*)(br + k + 0);
            a0 = __builtin_amdgcn_wmma_f32_16x16x4_f32(false, av, false, bv, (short)0, a0, false, false);
            av = *(const v2f*)(ar + k + 4);
            bv = *(const v2f*)(br + k + 4);
            a1 = __builtin_amdgcn_wmma_f32_16x16x4_f32(false, av, false, bv, (short)0, a1, false, false);
            av = *(const v2f*)(ar + k + 8);
            bv = *(const v2f*)(br + k + 8);
            a2 = __builtin_amdgcn_wmma_f32_16x16x4_f32(false, av, false, bv, (short)0, a2, false, false);
            av = *(const v2f*)(ar + k + 12);
            bv = *(const v2f*)(br + k + 12);
            a3 = __builtin_amdgcn_wmma_f32_16x16x4_f32(false, av, false, bv, (short)0, a3, false, false);
        }
        __syncthreads();
    }
#endif

    const v8f acc = (a0 + a1) + (a2 + a3);
    const int col = nbase + lr;
    const float bias = b_proj[col];

    #pragma unroll
    for (int r = 0; r < 8; ++r) {
        const int m = mbase + wave * 16 + r + 8 * lb;
        logits[(size_t)m * VOC + col] = acc[r] + bias;
    }
}

// ---------------------------------------------------------------------------
extern "C" void kernel_launch(void* const* d_in, const int* in_sizes, int n_in,
                              void* d_out, int out_size, void* d_ws, size_t ws_size,
                              hipStream_t stream) {
    const int*           tgt    = (const int*)d_in[0];
    const float*         enc    = (const float*)d_in[1];
    const float*         h0     = (const float*)d_in[2];
    // d_in[3] = c0 (unused by the reference math)
    const unsigned char* pad    = (const unsigned char*)d_in[4];
    // d_in[5] = max_len scalar (constant 50)
    const float*         emb    = (const float*)d_in[6];
    const float*         w_ih   = (const float*)d_in[7];
    const float*         w_hh   = (const float*)d_in[8];
    const float*         b_ih   = (const float*)d_in[9];
    const float*         b_hh   = (const float*)d_in[10];
    const float*         w_proj = (const float*)d_in[11];
    const float*         b_proj = (const float*)d_in[12];

    float* out    = (float*)d_out;
    float* logits = out;                                    // [B, V]
    float* h_out  = out + (size_t)BATCH * VOC;              // [1, B, E]
    float* c_out  = h_out + (size_t)BATCH * EDIM;           // [1, B, E]

    float* ws    = (float*)d_ws;
    float* words = ws;                                      // [B, E]
    float* ctx   = ws + (size_t)BATCH * EDIM;               // [B, E]

    attn_ctx_kernel<<<BATCH, 256, 0, stream>>>(tgt, enc, h0, pad, emb, words, ctx);

    // 4 m-blocks x 64 h-tiles = 256 blocks, 8 waves each
    gates_lstm_kernel<<<256, 256, 0, stream>>>(words, ctx, w_ih, w_hh,
                                               b_ih, b_hh, h_out, c_out);

    // 4 m-blocks x 3125 n-blocks = 12500 blocks, 8 waves each
    proj_kernel<<<12500, 256, 0, stream>>>(h_out, w_proj, b_proj, logits);
}